// MultiModal_39848706573683
// MI455X (gfx1250) — compile-verified
//
#include <hip/hip_runtime.h>
#include <math.h>

// ---------------------------------------------------------------------------
// MI455X (gfx1250) implementation. fp32 end-to-end (matches reference dtype).
// Heavy GEMMs -> V_WMMA_F32_16X16X4_F32, B-tiles via Tensor Data Mover.
// ---------------------------------------------------------------------------

typedef __attribute__((ext_vector_type(2))) float        v2f;
typedef __attribute__((ext_vector_type(8))) float        v8f;
typedef __attribute__((ext_vector_type(4))) unsigned int u32x4;
typedef __attribute__((ext_vector_type(8))) int          i32x8;
typedef __attribute__((ext_vector_type(4))) int          i32x4;

#define KT 16   // K slab per LDS stage (4 WMMA k-steps)

__device__ __forceinline__ float gelu_tanh(float x) {
  // jax.nn.gelu (approximate=True)
  return 0.5f * x * (1.0f + tanhf(0.7978845608028654f * (x + 0.044715f * x * x * x)));
}

// ---------------------------------------------------------------------------
// Generic WMMA f32 GEMM:  C = act( (A[M,K] @ B[K,N] + bias) * (g*rsqrt(1+eps)) + be )
// A,B,C row-major. 256 threads = 8 wave32 waves; each wave: WTMx WTN 16x16 tiles.
// Batched along blockIdx.z with element strides sA/sB/sC/sBias.
// ACT: 0=none 1=relu 2=gelu.  BNE: batchnorm affine enabled.  TDM: B tile by DMA.
// ---------------------------------------------------------------------------
template<int BM, int BN, int WROWS, int WCOLS, int WTM, int WTN, int ACT, bool BNE, bool TDM>
__global__ __launch_bounds__(256) void gemm_wmma(
    const float* __restrict__ Ab, const float* __restrict__ Bb, float* __restrict__ Cb,
    const float* __restrict__ biasB, const float* __restrict__ gammaP, const float* __restrict__ betaP,
    int M, int N, int K,
    long long sA, long long sB, long long sC, long long sBias)
{
  constexpr int AS = KT + 4;   // As row stride (20 dwords -> conflict-free b64 frag reads)
  constexpr int BS = BN + 8;   // Bs row stride (2*BS % 64 == 16 -> half-waves hit disjoint banks)
  __shared__ float As[BM * AS];
  __shared__ float Bs[KT * BS];

  const float* A    = Ab + (size_t)blockIdx.z * sA;
  const float* B    = Bb + (size_t)blockIdx.z * sB;
  float*       C    = Cb + (size_t)blockIdx.z * sC;
  const float* bias = biasB ? (biasB + (size_t)blockIdx.z * sBias) : nullptr;

  const int tid  = threadIdx.x;
  const int wid  = tid >> 5;
  const int lane = tid & 31;
  const int lrow = lane & 15;
  const int lhi  = lane >> 4;

  const int bm = blockIdx.y * BM;
  const int bn = blockIdx.x * BN;
  const int wr = wid / WCOLS, wc = wid % WCOLS;
  const int wm = wr * (WTM * 16);
  const int wn = wc * (WTN * 16);

  v8f acc[WTM][WTN];
#pragma unroll
  for (int i = 0; i < WTM; i++)
#pragma unroll
    for (int j = 0; j < WTN; j++)
#pragma unroll
      for (int r = 0; r < 8; r++) acc[i][j][r] = 0.0f;

  for (int k0 = 0; k0 < K; k0 += KT) {
    __syncthreads();

    if (TDM) {
      // ---- Tensor Data Mover: B tile (KTxBN) -> Bs, padded rows (pad 8 dwords / 128)
      if (wid == 0) {
        unsigned long long ga = (unsigned long long)(const void*)(B + (size_t)k0 * N + bn);
        unsigned ldsOff = (unsigned)(unsigned long long)(void*)&Bs[0];
        u32x4 g0;
        g0[0] = 1u;                                       // count=1, user mode
        g0[1] = ldsOff;                                   // lds_addr
        g0[2] = (unsigned)(ga & 0xFFFFFFFFull);           // global_addr lo
        g0[3] = (unsigned)((ga >> 32) & 0x1FFFFFFull) | (2u << 30);  // addr hi | type=2
        unsigned td0 = (unsigned)(N - bn);                // remaining cols (OOB -> zero fill)
        unsigned td1 = (unsigned)(K - k0);                // remaining rows
        i32x8 g1;
        g1[0] = (int)((2u << 16)  |                       // data_size = 4B
                      (1u << 20)  |                       // pad_enable
                      (6u << 22)  |                       // pad_interval = 128 dwords
                      (7u << 25));                        // pad_amount   = 8 dwords
        g1[1] = (int)((td0 & 0xFFFFu) << 16);             // tensor_dim0[15:0]
        g1[2] = (int)((td0 >> 16) | ((td1 & 0xFFFFu) << 16));
        g1[3] = (int)((td1 >> 16) | ((unsigned)BN << 16));// tile_dim0 = BN
        g1[4] = (int)KT;                                  // tile_dim1 = KT, tile_dim2 = 0
        g1[5] = (int)(unsigned)N;                         // tensor_dim0_stride lo
        g1[6] = 0;
        g1[7] = 0;
        i32x4 z4 = {0, 0, 0, 0};
        i32x8 z8 = {0, 0, 0, 0, 0, 0, 0, 0};
        // 6-arg toolchain form (clang-23 / therock headers)
        __builtin_amdgcn_tensor_load_to_lds(g0, g1, z4, z4, z8, 0);
      }
    } else {
#pragma unroll
      for (int i = 0; i < (KT * BN) / 256; ++i) {
        int idx = tid + 256 * i;
        int r = idx / BN, c = idx % BN;
        int gk = k0 + r, gn = bn + c;
        Bs[r * BS + c] = (gk < K && gn < N) ? B[(size_t)gk * N + gn] : 0.0f;
      }
    }

    // ---- A tile (BMxKT) -> As (all waves)
#pragma unroll
    for (int i = 0; i < (BM * KT) / 256; ++i) {
      int idx = tid + 256 * i;
      int r = idx / KT, c = idx % KT;
      int gm = bm + r, gk = k0 + c;
      As[r * AS + c] = (gm < M && gk < K) ? A[(size_t)gm * K + gk] : 0.0f;
    }
    // hint next A slab into cache
    if (k0 + KT < K) {
      int r = tid / KT, c = tid % KT;
      int gm = bm + r, gk = k0 + KT + c;
      if (gm < M && gk < K) __builtin_prefetch(&A[(size_t)gm * K + gk], 0, 3);
    }

    if (TDM) { if (wid == 0) __builtin_amdgcn_s_wait_tensorcnt(0); }
    __syncthreads();

    // ---- compute: 4 k-steps of 16x16x4 WMMA per slab
#pragma unroll
    for (int kk = 0; kk < KT; kk += 4) {
      v2f af[WTM];
      v2f bf[WTN];
#pragma unroll
      for (int i = 0; i < WTM; i++) {
        // A frag 16x4 f32: lanes 0-15 -> K=kk,kk+1 ; lanes 16-31 -> K=kk+2,kk+3
        int row = wm + i * 16 + lrow;
        int col = kk + 2 * lhi;
        af[i].x = As[row * AS + col];
        af[i].y = As[row * AS + col + 1];
      }
#pragma unroll
      for (int j = 0; j < WTN; j++) {
        // B frag 4x16 f32: VGPR v holds K = kk + 2*lhi + v, N = lane&15
        int colB = wn + j * 16 + lrow;
        int r0 = kk + 2 * lhi;
        bf[j].x = Bs[r0 * BS + colB];
        bf[j].y = Bs[(r0 + 1) * BS + colB];
      }
#pragma unroll
      for (int i = 0; i < WTM; i++)
#pragma unroll
        for (int j = 0; j < WTN; j++)
          acc[i][j] = __builtin_amdgcn_wmma_f32_16x16x4_f32(
              false, af[i], false, bf[j], (short)0, acc[i][j], false, false);
    }
  }

  // ---- epilogue + store (C/D layout: VGPR r -> M = r + 8*lhi, N = lane&15)
  const float bnscale = rsqrtf(1.0f + 1e-5f);
#pragma unroll
  for (int i = 0; i < WTM; i++) {
#pragma unroll
    for (int j = 0; j < WTN; j++) {
      const int colg = bn + wn + j * 16 + lrow;
      if (colg < N) {
        float bv = bias ? bias[colg] : 0.0f;
        float sc = 1.0f, sh = 0.0f;
        if (BNE) { sc = gammaP[colg] * bnscale; sh = betaP[colg]; }
#pragma unroll
        for (int r = 0; r < 8; r++) {
          const int rowg = bm + wm + i * 16 + r + 8 * lhi;
          if (rowg < M) {
            float v = acc[i][j][r] + bv;
            if (BNE) v = v * sc + sh;
            if (ACT == 1) v = fmaxf(v, 0.0f);
            if (ACT == 2) v = gelu_tanh(v);
            C[(size_t)rowg * N + colg] = v;
          }
        }
      }
    }
  }
}

// ---------------------------------------------------------------------------
// Small kernels
// ---------------------------------------------------------------------------
#define BATCH 512
#define HID   768
#define LH    (64 * 768)   // L*H stride inside all_hidden_states per batch row

// per-layer sum over (B,H) of CLS token, scaled by weights13
__global__ __launch_bounds__(256) void k_sum13(const float* __restrict__ hs,
                                               const float* __restrict__ w13,
                                               float* __restrict__ sums) {
  __shared__ float red[256];
  const int l = blockIdx.x, t = threadIdx.x;
  const float* base = hs + (size_t)l * BATCH * LH;
  float s = 0.0f;
  for (int i = t; i < BATCH * HID; i += 256) {
    int b = i / HID, h = i % HID;
    s += base[(size_t)b * LH + h];
  }
  red[t] = s; __syncthreads();
  for (int st = 128; st > 0; st >>= 1) { if (t < st) red[t] += red[t + st]; __syncthreads(); }
  if (t == 0) sums[l] = red[0] * w13[l];
}

__global__ void k_softmax13(const float* __restrict__ s, float* __restrict__ a) {
  if (threadIdx.x == 0 && blockIdx.x == 0) {
    float mx = s[0];
    for (int l = 1; l < 13; l++) mx = fmaxf(mx, s[l]);
    float e[13], sum = 0.0f;
    for (int l = 0; l < 13; l++) { e[l] = expf(s[l] - mx); sum += e[l]; }
    for (int l = 0; l < 13; l++) a[l] = e[l] / sum;
  }
}

// text_raw[b,h] = sum_l atten[l] * hs[l,b,0,h]
__global__ __launch_bounds__(256) void k_textraw(const float* __restrict__ hs,
                                                 const float* __restrict__ atten,
                                                 float* __restrict__ out) {
  __shared__ float a[13];
  if (threadIdx.x < 13) a[threadIdx.x] = atten[threadIdx.x];
  __syncthreads();
  int idx = blockIdx.x * 256 + threadIdx.x;   // exact grid: B*H
  int b = idx / HID, h = idx % HID;
  float s = 0.0f;
  for (int l = 0; l < 13; l++)
    s += a[l] * hs[((size_t)l * BATCH + b) * (size_t)LH + h];
  out[idx] = s;
}

// dst[b, 0:w1] = s1[b], dst[b, w1:w1+w2] = s2[b]
__global__ void k_concat2(float* __restrict__ dst, const float* __restrict__ s1, int w1,
                          const float* __restrict__ s2, int w2) {
  int w = w1 + w2, total = BATCH * w;
  for (int i = blockIdx.x * blockDim.x + threadIdx.x; i < total; i += gridDim.x * blockDim.x) {
    int b = i / w, c = i % w;
    dst[i] = (c < w1) ? s1[(size_t)b * w1 + c] : s2[(size_t)b * w2 + (c - w1)];
  }
}

// GAT: leaky-relu attention over 5 entity nodes + ELU; one block per batch row
__global__ __launch_bounds__(256) void k_gat(const float* __restrict__ tfeat,
                                             const float* __restrict__ efeat,
                                             const float* __restrict__ ga,
                                             float* __restrict__ zed) {
  __shared__ float red[256];
  __shared__ float dots[6];
  __shared__ float attn[5];
  const int b = blockIdx.x, t = threadIdx.x;
  const float* tf = tfeat + (size_t)b * HID;
  for (int n = 0; n < 6; n++) {
    float p = 0.0f;
    if (n == 0) {
      for (int h = t; h < HID; h += 256) p += tf[h] * ga[h];
    } else {
      const float* ef = efeat + ((size_t)b * 5 + (n - 1)) * HID;
      for (int h = t; h < HID; h += 256) p += ef[h] * ga[HID + h];
    }
    red[t] = p; __syncthreads();
    for (int st = 128; st > 0; st >>= 1) { if (t < st) red[t] += red[t + st]; __syncthreads(); }
    if (t == 0) dots[n] = red[0];
    __syncthreads();
  }
  if (t == 0) {
    float e[5];
    for (int n = 0; n < 5; n++) { float x = dots[0] + dots[n + 1]; e[n] = (x > 0.f) ? x : 0.2f * x; }
    float mx = e[0];
    for (int n = 1; n < 5; n++) mx = fmaxf(mx, e[n]);
    float s = 0.0f;
    for (int n = 0; n < 5; n++) { e[n] = expf(e[n] - mx); s += e[n]; }
    for (int n = 0; n < 5; n++) attn[n] = e[n] / s;
  }
  __syncthreads();
  for (int h = t; h < HID; h += 256) {
    float v = tf[h];
    for (int n = 0; n < 5; n++) v += attn[n] * efeat[((size_t)b * 5 + n) * HID + h];
    zed[(size_t)b * HID + h] = (v > 0.f) ? v : (expf(v) - 1.0f);
  }
}

// cosine-sim * w + b, then scale correlation rows in place; one wave32 per batch row
__global__ __launch_bounds__(256) void k_sim(const float* __restrict__ text,
                                             const float* __restrict__ image,
                                             const float* __restrict__ wS, const float* __restrict__ bS,
                                             float* __restrict__ corr) {
  int wid = threadIdx.x >> 5, lane = threadIdx.x & 31;
  int b = blockIdx.x * 8 + wid;
  const float* tp = text + (size_t)b * 512;
  const float* ip = image + (size_t)b * 512;
  float sti = 0.f, stt = 0.f, sii = 0.f;
  for (int k = lane; k < 512; k += 32) {
    float tv = tp[k], iv = ip[k];
    sti += tv * iv; stt += tv * tv; sii += iv * iv;
  }
  for (int off = 16; off > 0; off >>= 1) {
    sti += __shfl_down(sti, off, 32);
    stt += __shfl_down(stt, off, 32);
    sii += __shfl_down(sii, off, 32);
  }
  float sim = __shfl(sti, 0, 32) / (sqrtf(__shfl(stt, 0, 32)) * sqrtf(__shfl(sii, 0, 32)));
  sim = sim * wS[0] + bS[0];
  float* cr = corr + (size_t)b * 64;
  for (int d = lane; d < 64; d += 32) cr[d] *= sim;
}

// softmax gate over E=4 experts; one thread per batch row
__global__ void k_gate(const float* __restrict__ x, const float* __restrict__ w,
                       const float* __restrict__ bb, float* __restrict__ g) {
  int b = blockIdx.x * blockDim.x + threadIdx.x;
  if (b >= BATCH) return;
  const float* xp = x + (size_t)b * 64;
  float lg[4];
  for (int e = 0; e < 4; e++) {
    float s = bb[e];
    for (int d = 0; d < 64; d++) s += xp[d] * w[d * 4 + e];
    lg[e] = s;
  }
  float mx = fmaxf(fmaxf(lg[0], lg[1]), fmaxf(lg[2], lg[3]));
  float sum = 0.f;
  for (int e = 0; e < 4; e++) { lg[e] = expf(lg[e] - mx); sum += lg[e]; }
  for (int e = 0; e < 4; e++) g[(size_t)b * 4 + e] = lg[e] / sum;
}

// fold (h+pos)@W+b  ->  h@W + (pos@W+b):  pw[e*2+j][h] = sum_d pos[d]*ew[e,j,d,h] + eb[e,j,h]
__global__ void k_posw(const float* __restrict__ pos, const float* __restrict__ ew,
                       const float* __restrict__ eb, float* __restrict__ pw) {
  int i = blockIdx.x * blockDim.x + threadIdx.x;
  if (i >= 8 * 64) return;
  int h = i & 63, ej = i >> 6;
  const float* W = ew + (size_t)ej * 64 * 64;
  float s = eb[ej * 64 + h];
  for (int d = 0; d < 64; d++) s += pos[d] * W[d * 64 + h];
  pw[i] = s;
}

// out[b,d] = sum_e gate[b,e] * H[e,b,d]
__global__ void k_comb(const float* __restrict__ H, const float* __restrict__ gate,
                       float* __restrict__ out) {
  int i = blockIdx.x * 256 + threadIdx.x;   // exact grid: B*64
  int b = i >> 6;
  float s = 0.f;
  for (int e = 0; e < 4; e++) s += gate[(size_t)b * 4 + e] * H[(size_t)e * BATCH * 64 + i];
  out[i] = s;
}

// 3-token attention pooling; one thread per batch row
__global__ void k_pool(const float* __restrict__ si, const float* __restrict__ st,
                       const float* __restrict__ z64, const float* __restrict__ aw,
                       float* __restrict__ pooled) {
  int b = blockIdx.x * blockDim.x + threadIdx.x;
  if (b >= BATCH) return;
  const float* f0 = si + (size_t)b * 64;
  const float* f1 = st + (size_t)b * 64;
  const float* f2 = z64 + (size_t)b * 64;
  float sc[3];
  const float* fp[3] = {f0, f1, f2};
  for (int t = 0; t < 3; t++) {
    float s = 0.f;
    for (int d = 0; d < 64; d++) s += fp[t][d] * aw[d];
    sc[t] = s;
  }
  float mx = fmaxf(sc[0], fmaxf(sc[1], sc[2]));
  float sum = 0.f;
  for (int t = 0; t < 3; t++) { sc[t] = expf(sc[t] - mx); sum += sc[t]; }
  for (int t = 0; t < 3; t++) sc[t] /= sum;
  for (int d = 0; d < 64; d++)
    pooled[(size_t)b * 64 + d] = sc[0] * f0[d] + sc[1] * f1[d] + sc[2] * f2[d];
}

__global__ void k_add(float* __restrict__ a, const float* __restrict__ b2, int n) {
  int i = blockIdx.x * blockDim.x + threadIdx.x;
  if (i < n) a[i] += b2[i];
}

// 64 -> 2 classifier head; one thread per (b, c)
__global__ void k_cls(const float* __restrict__ lite, const float* __restrict__ w,
                      const float* __restrict__ bb, float* __restrict__ out) {
  int i = blockIdx.x * blockDim.x + threadIdx.x;
  if (i >= BATCH * 2) return;
  int b = i >> 1, c = i & 1;
  float s = bb[c];
  const float* lp = lite + (size_t)b * 64;
  for (int d = 0; d < 64; d++) s += lp[d] * w[d * 2 + c];
  out[i] = s;
}

// ---------------------------------------------------------------------------
// Host launcher
// ---------------------------------------------------------------------------
extern "C" void kernel_launch(void* const* d_in, const int* in_sizes, int n_in,
                              void* d_out, int out_size, void* d_ws, size_t ws_size,
                              hipStream_t stream) {
  (void)in_sizes; (void)n_in; (void)out_size; (void)ws_size;
  auto F = [&](int i) { return (const float*)d_in[i]; };

  // inputs (setup_inputs dict order; params dict flattened in insertion order)
  const float* hs        = F(1);
  const float* image_raw = F(2);
  const float* text      = F(3);
  const float* image     = F(4);
  const float* er        = F(5);
  const float* w13   = F(6);
  const float* gat_W = F(7);
  const float* gat_a = F(8);
  const float* er_w = F(9),  *er_b = F(10), *er_g = F(11), *er_be = F(12);
  const float* tu_w1 = F(13), *tu_b1 = F(14), *tu_g1 = F(15), *tu_be1 = F(16);
  const float* tu_w2 = F(17), *tu_b2 = F(18), *tu_g2 = F(19), *tu_be2 = F(20);
  const float* iu_w1 = F(21), *iu_b1 = F(22), *iu_g1 = F(23), *iu_be1 = F(24);
  const float* iu_w2 = F(25), *iu_b2 = F(26), *iu_g2 = F(27), *iu_be2 = F(28);
  const float* cm_w1 = F(29), *cm_b1 = F(30), *cm_g1 = F(31), *cm_be1 = F(32);
  const float* cm_w2 = F(33), *cm_b2 = F(34), *cm_g2 = F(35), *cm_be2 = F(36);
  const float* wS = F(37), *bS = F(38);
  const float* pos_img = F(39), *pos_txt = F(40), *pos_fus = F(41);
  const float* img_ew = F(42), *img_eb = F(43);
  const float* txt_ew = F(44), *txt_eb = F(45);
  const float* fus_ew = F(46), *fus_eb = F(47);
  const float* gimg_w = F(48), *gimg_b = F(49);
  const float* gtxt_w = F(50), *gtxt_b = F(51);
  const float* gmain_w = F(52), *gmain_b = F(53);
  const float* attn_w = F(54);
  const float* zed_w = F(55), *zed_b = F(56);
  const float* imtr_w = F(57), *imtr_b = F(58);
  const float* txtr_w = F(59), *txtr_b = F(60);
  const float* mxtr_w = F(61), *mxtr_b = F(62);
  const float* imcl_w = F(63), *imcl_b = F(64);
  const float* txcl_w = F(65), *txcl_b = F(66);
  const float* mxcl_w = F(67), *mxcl_b = F(68);

  // workspace carve-up (floats)
  float* ws = (float*)d_ws;
  size_t off = 0;
  auto alloc = [&](size_t n) { float* p = ws + off; off += (n + 15) & ~(size_t)15; return p; };
  float* sums      = alloc(13);
  float* atten     = alloc(13);
  float* text_raw  = alloc(512 * 768);
  float* er_p      = alloc(512 * 3840);
  float* tfeat     = alloc(512 * 768);
  float* efeat     = alloc(512 * 5 * 768);
  float* zed       = alloc(512 * 768);
  float* cat_tu    = alloc(512 * 1280);
  float* cat_iu    = alloc(512 * 1512);
  float* cat_cm    = alloc(512 * 1024);
  float* h_tu      = alloc(512 * 256);
  float* h_iu      = alloc(512 * 256);
  float* h_cm      = alloc(512 * 256);
  float* textp     = alloc(512 * 64);
  float* imagep    = alloc(512 * 64);
  float* corr      = alloc(512 * 64);
  float* gate_img  = alloc(512 * 4);
  float* gate_txt  = alloc(512 * 4);
  float* gate_main = alloc(512 * 4);
  float* posw_img  = alloc(8 * 64);
  float* posw_txt  = alloc(8 * 64);
  float* posw_fus  = alloc(8 * 64);
  float* Hb0       = alloc(4 * 512 * 64);
  float* Hb1       = alloc(4 * 512 * 64);
  float* sh_img    = alloc(512 * 64);
  float* sh_txt    = alloc(512 * 64);
  float* zed64     = alloc(512 * 64);
  float* pooled    = alloc(512 * 64);
  float* finalf    = alloc(512 * 64);
  float* lite      = alloc(512 * 64);

  float* outF = (float*)d_out;   // [mix(512x2) | image_only(512x2) | text_only(512x2)]

  // GEMM wrappers -------------------------------------------------------------
  auto gemmBigBnRelu = [&](const float* A, const float* B, float* C,
                           const float* bi, const float* g, const float* be,
                           int M, int N, int K) {
    dim3 gr((N + 127) / 128, (M + 63) / 64, 1);
    gemm_wmma<64, 128, 2, 4, 2, 2, 1, true, true><<<gr, 256, 0, stream>>>(
        A, B, C, bi, g, be, M, N, K, 0, 0, 0, 0);
  };
  auto gemmBigPlain = [&](const float* A, const float* B, float* C, int M, int N, int K) {
    dim3 gr((N + 127) / 128, (M + 63) / 64, 1);
    gemm_wmma<64, 128, 2, 4, 2, 2, 0, false, true><<<gr, 256, 0, stream>>>(
        A, B, C, nullptr, nullptr, nullptr, M, N, K, 0, 0, 0, 0);
  };
  auto gemmNBnRelu = [&](const float* A, const float* B, float* C,
                         const float* bi, const float* g, const float* be,
                         int M, int N, int K) {
    dim3 gr((N + 63) / 64, (M + 127) / 128, 1);
    gemm_wmma<128, 64, 4, 2, 2, 2, 1, true, false><<<gr, 256, 0, stream>>>(
        A, B, C, bi, g, be, M, N, K, 0, 0, 0, 0);
  };
  auto gemmNBias = [&](const float* A, const float* B, float* C, const float* bi,
                       int M, int N, int K) {
    dim3 gr((N + 63) / 64, (M + 127) / 128, 1);
    gemm_wmma<128, 64, 4, 2, 2, 2, 0, false, false><<<gr, 256, 0, stream>>>(
        A, B, C, bi, nullptr, nullptr, M, N, K, 0, 0, 0, 0);
  };
  auto gemmNBiasRelu = [&](const float* A, const float* B, float* C, const float* bi,
                           int M, int N, int K) {
    dim3 gr((N + 63) / 64, (M + 127) / 128, 1);
    gemm_wmma<128, 64, 4, 2, 2, 2, 1, false, false><<<gr, 256, 0, stream>>>(
        A, B, C, bi, nullptr, nullptr, M, N, K, 0, 0, 0, 0);
  };
  auto gemmExpertLayer = [&](const float* A, long long sA, const float* B, long long sB,
                             float* C, long long sC, const float* bi, long long sBias) {
    dim3 gr(1, 4, 4);  // N=64, M=512, batch=E=4
    gemm_wmma<128, 64, 4, 2, 2, 2, 2, false, false><<<gr, 256, 0, stream>>>(
        A, B, C, bi, nullptr, nullptr, 512, 64, 64, sA, sB, sC, sBias);
  };
  auto experts = [&](const float* X, const float* ew, const float* posw,
                     const float* gate, float* out) {
    // layer 0: A shared across experts (sA=0); ew[e][0] stride 2*64*64; bias posw[e*2+0]
    gemmExpertLayer(X, 0, ew, 2 * 64 * 64, Hb0, 512 * 64, posw, 2 * 64);
    // layer 1
    gemmExpertLayer(Hb0, 512 * 64, ew + 64 * 64, 2 * 64 * 64, Hb1, 512 * 64, posw + 64, 2 * 64);
    k_comb<<<128, 256, 0, stream>>>(Hb1, gate, out);
  };

  // ---- pipeline -------------------------------------------------------------
  // 1) 13-layer CLS pooling -> atten -> text_raw
  k_sum13<<<13, 256, 0, stream>>>(hs, w13, sums);
  k_softmax13<<<1, 1, 0, stream>>>(sums, atten);
  k_textraw<<<(512 * 768) / 256, 256, 0, stream>>>(hs, atten, text_raw);

  // 2) big entity-relation GEMM + BN + ReLU (512x3840x3840)
  gemmBigBnRelu(er, er_w, er_p, er_b, er_g, er_be, 512, 3840, 3840);

  // 3) concat inputs for the three MLP2s
  k_concat2<<<(512 * 1280) / 256, 256, 0, stream>>>(cat_tu, text_raw, 768, text, 512);
  k_concat2<<<(512 * 1512) / 256, 256, 0, stream>>>(cat_iu, image_raw, 1000, image, 512);
  k_concat2<<<(512 * 1024) / 256, 256, 0, stream>>>(cat_cm, text, 512, image, 512);

  // 4) GAT projections + attention + ELU -> zed
  gemmBigPlain(text_raw, gat_W, tfeat, 512, 768, 768);
  gemmBigPlain(er_p, gat_W, efeat, 2560, 768, 768);   // er_feats_in = er_p.reshape(B,5,768)
  k_gat<<<512, 256, 0, stream>>>(tfeat, efeat, gat_a, zed);

  // 5) MLP2 stacks (w1: big config; w2: narrow N=64 config)
  gemmBigBnRelu(cat_tu, tu_w1, h_tu, tu_b1, tu_g1, tu_be1, 512, 256, 1280);
  gemmBigBnRelu(cat_iu, iu_w1, h_iu, iu_b1, iu_g1, iu_be1, 512, 256, 1512);
  gemmBigBnRelu(cat_cm, cm_w1, h_cm, cm_b1, cm_g1, cm_be1, 512, 256, 1024);
  gemmNBnRelu(h_tu, tu_w2, textp, tu_b2, tu_g2, tu_be2, 512, 64, 256);
  gemmNBnRelu(h_iu, iu_w2, imagep, iu_b2, iu_g2, iu_be2, 512, 64, 256);
  gemmNBnRelu(h_cm, cm_w2, corr, cm_b2, cm_g2, cm_be2, 512, 64, 256);

  // 6) cosine-sim scaling of correlation
  k_sim<<<64, 256, 0, stream>>>(text, image, wS, bS, corr);

  // 7) gates + folded pos biases + per-modality expert MoE
  k_gate<<<2, 256, 0, stream>>>(imagep, gimg_w, gimg_b, gate_img);
  k_gate<<<2, 256, 0, stream>>>(textp, gtxt_w, gtxt_b, gate_txt);
  k_posw<<<2, 256, 0, stream>>>(pos_img, img_ew, img_eb, posw_img);
  k_posw<<<2, 256, 0, stream>>>(pos_txt, txt_ew, txt_eb, posw_txt);
  k_posw<<<2, 256, 0, stream>>>(pos_fus, fus_ew, fus_eb, posw_fus);
  experts(imagep, img_ew, posw_img, gate_img, sh_img);
  experts(textp, txt_ew, posw_txt, gate_txt, sh_txt);

  // 8) zed64, 3-token attention pooling, main gate, fuse experts (token 0 only)
  gemmNBias(zed, zed_w, zed64, zed_b, 512, 64, 768);
  k_pool<<<2, 256, 0, stream>>>(sh_img, sh_txt, zed64, attn_w, pooled);
  k_gate<<<2, 256, 0, stream>>>(pooled, gmain_w, gmain_b, gate_main);
  experts(sh_img, fus_ew, posw_fus, gate_main, finalf);   // concat_feat[:,0] == shared_img
  k_add<<<128, 256, 0, stream>>>(finalf, corr, 512 * 64);

  // 9) trims + classifier heads
  gemmNBiasRelu(sh_img, imtr_w, lite, imtr_b, 512, 64, 64);
  k_cls<<<4, 256, 0, stream>>>(lite, imcl_w, imcl_b, outF + 512 * 2);
  gemmNBiasRelu(sh_txt, txtr_w, lite, txtr_b, 512, 64, 64);
  k_cls<<<4, 256, 0, stream>>>(lite, txcl_w, txcl_b, outF + 2 * 512 * 2);
  gemmNBiasRelu(finalf, mxtr_w, lite, mxtr_b, 512, 64, 64);
  k_cls<<<4, 256, 0, stream>>>(lite, mxcl_w, mxcl_b, outF);
}